// SDDMMNet_17961553232201
// MI455X (gfx1250) — compile-verified
//
#include <hip/hip_runtime.h>
#include <stdint.h>

// SDDMM: out[M,N] = x[M,N] * (y[M,K] @ z[K,N]);  M=N=8192, K=64, f32.
// Memory-bound (~516 MB @ 23.3 TB/s ~= 22us). Strategy:
//  - z slab held persistently in VGPRs as WMMA B-fragments (reused over all M)
//  - y tiles staged by the Tensor Data Mover into double-buffered LDS
//  - x / out streamed once, coalesced, laid out to match WMMA C/D lanes
//  - full f32 precision via V_WMMA_F32_16X16X4_F32 (K=4 per issue, 16 issues
//    cover K=64) -- f32 matrix rate exceeds the ~390 TFLOP/s needed
//  - A-frags preloaded into a register array so the 32-WMMA stream is not
//    punctuated by per-pair s_wait_dscnt stalls.

typedef __attribute__((ext_vector_type(2))) float v2f;
typedef __attribute__((ext_vector_type(8))) float v8f;
typedef __attribute__((ext_vector_type(4))) unsigned int u32x4;
typedef __attribute__((ext_vector_type(4))) int i32x4;
typedef __attribute__((ext_vector_type(8))) int i32x8;

#define M_DIM 8192
#define N_DIM 8192
#define K_DIM 64
#define BLOCK_N 256        // 8 waves * 32 columns
#define M_CHUNK 256        // rows per block
#define ROWS_PER_ITER 16   // one 16-row WMMA tile strip per iteration
#define ITERS (M_CHUNK / ROWS_PER_ITER)
#define Y_TILE_BYTES (ROWS_PER_ITER * K_DIM * 4)   // 4 KB

// Issue one TDM load of a 16x64 f32 tile of y into LDS.
// D# per CDNA5 ISA:
//   group0: count=1, lds_addr, global_addr[56:0], type=2
//   group1: data_size=4B(code 2), tensor_dim0=64, tensor_dim1=8192,
//           tile_dim0=64, tile_dim1=16, tensor_dim0_stride=64
//   groups 2/3 (and the trailing group): zero (2D tensor)
__device__ __forceinline__ void tdm_load_y_tile(const float* yg, int row0,
                                                unsigned lds_off) {
  unsigned long long ga =
      (unsigned long long)(uintptr_t)(yg + (size_t)row0 * K_DIM);
  u32x4 g0;
  g0[0] = 1u;                                     // count=1 (valid user D#)
  g0[1] = lds_off;                                // lds_addr (bytes)
  g0[2] = (unsigned)(ga & 0xFFFFFFFFull);         // global_addr[31:0]
  g0[3] = (unsigned)((ga >> 32) & 0x01FFFFFFull)  // global_addr[56:32]
          | (2u << 30);                           // type=2 ("image")
  i32x8 g1;
  g1[0] = (int)(2u << 16);                        // data_size = 4 bytes
  g1[1] = (int)((unsigned)K_DIM << 16);           // tensor_dim0[15:0] = 64
  g1[2] = (int)(((unsigned)K_DIM >> 16) |
                (((unsigned)M_DIM & 0xFFFFu) << 16)); // tensor_dim1[15:0]
  g1[3] = (int)(((unsigned)M_DIM >> 16) |
                ((unsigned)K_DIM << 16));         // tile_dim0 = 64
  g1[4] = (int)(unsigned)ROWS_PER_ITER;           // tile_dim1=16, tile_dim2=0
  g1[5] = (int)K_DIM;                             // tensor_dim0_stride = 64
  g1[6] = 0;                                      // stride0 hi / stride1 lo
  g1[7] = 0;
  i32x4 zero4 = {0, 0, 0, 0};
  i32x8 zero8 = {0, 0, 0, 0, 0, 0, 0, 0};
  // amdgpu-toolchain (clang-23) 6-arg form:
  // (u32x4 g0, i32x8 g1, i32x4 g2, i32x4 g3, i32x8 extra, i32 cpol)
  __builtin_amdgcn_tensor_load_to_lds(g0, g1, zero4, zero4, zero8, 0);
}

__global__ __launch_bounds__(256) void sddmm_wmma_kernel(
    const float* __restrict__ x, const float* __restrict__ y,
    const float* __restrict__ z, float* __restrict__ out) {
  __shared__ float sY[2][ROWS_PER_ITER * K_DIM];   // double-buffered y tiles

  const int tid    = threadIdx.x;
  const int wave   = tid >> 5;
  const int lane   = tid & 31;
  const int lane16 = lane & 15;
  const int half   = lane >> 4;     // 0: lanes 0-15, 1: lanes 16-31

  const int cbase  = blockIdx.x * BLOCK_N + wave * 32;  // wave's 32 columns
  const int rowblk = blockIdx.y * M_CHUNK;

  // ---- Persistent B fragments: z[64 x 32] for this wave, in VGPRs. ----
  // 4x16 f32 B tile layout per ISA: VGPR0 = rows {4s, 4s+2} (lane halves),
  // VGPR1 = rows {4s+1, 4s+3}; columns striped across lanes 0-15.
  v2f b[2][16];
#pragma unroll
  for (int t = 0; t < 2; ++t) {
#pragma unroll
    for (int s = 0; s < 16; ++s) {
      const int krow = 4 * s + 2 * half;
      const float* zp = z + (size_t)krow * N_DIM + (cbase + t * 16 + lane16);
      v2f bv;
      bv.x = zp[0];
      bv.y = zp[N_DIM];
      b[t][s] = bv;
    }
  }

  const unsigned lds_base = (unsigned)(uintptr_t)(void*)&sY[0][0];

  // Prologue: wave 0 kicks off the TDM load for iteration 0.
  if (wave == 0) tdm_load_y_tile(y, rowblk, lds_base);

  for (int i = 0; i < ITERS; ++i) {
    if (wave == 0) {
      if (i + 1 < ITERS) {
        // Issue next tile, then wait for the older one (pipeline depth 2).
        tdm_load_y_tile(y, rowblk + (i + 1) * ROWS_PER_ITER,
                        lds_base + (unsigned)(((i + 1) & 1) * Y_TILE_BYTES));
        __builtin_amdgcn_s_wait_tensorcnt(1);
      } else {
        __builtin_amdgcn_s_wait_tensorcnt(0);
      }
    }
    __syncthreads();   // LDS buffer (i&1) now valid for all 8 waves

    const int row0 = rowblk + i * ROWS_PER_ITER;

    // Prefetch x tile (issued early; outstanding during the WMMA loop).
    // C/D layout: VGPR v, lane l -> row = v + 8*half, col = lane16.
    float xv[2][8];
#pragma unroll
    for (int t = 0; t < 2; ++t) {
      const int col = cbase + t * 16 + lane16;
#pragma unroll
      for (int v = 0; v < 8; ++v) {
        const int row = row0 + v + 8 * half;
        xv[t][v] = x[(size_t)row * N_DIM + col];
      }
    }

    // Preload ALL A-frags for this 16-row strip into registers first:
    // 8x ds_load_2addr_b64 issued back-to-back, one s_wait_dscnt, then an
    // uninterrupted stream of 32 WMMAs.
    // A-frag (16x4 f32): lane l holds row lane16, K = 4s + 2*half + {0,1}.
    const float* sA = &sY[i & 1][0];
    v2f a[16];
#pragma unroll
    for (int s = 0; s < 16; ++s) {
      a[s] = *(const v2f*)(sA + lane16 * K_DIM + 4 * s + 2 * half);
    }

    v8f acc0 = {};
    v8f acc1 = {};
#pragma unroll
    for (int s = 0; s < 16; ++s) {
      acc0 = __builtin_amdgcn_wmma_f32_16x16x4_f32(
          false, a[s], false, b[0][s], (short)0, acc0, false, false);
      acc1 = __builtin_amdgcn_wmma_f32_16x16x4_f32(
          false, a[s], false, b[1][s], (short)0, acc1, false, false);
    }

    // Epilogue: out = x * acc, coalesced b32 stores.
#pragma unroll
    for (int t = 0; t < 2; ++t) {
      const v8f acc = t ? acc1 : acc0;
      const int col = cbase + t * 16 + lane16;
#pragma unroll
      for (int v = 0; v < 8; ++v) {
        const int row = row0 + v + 8 * half;
        out[(size_t)row * N_DIM + col] = xv[t][v] * acc[v];
      }
    }
    __syncthreads();   // everyone done with buffer (i&1) before reuse
  }
}

extern "C" void kernel_launch(void* const* d_in, const int* in_sizes, int n_in,
                              void* d_out, int out_size, void* d_ws,
                              size_t ws_size, hipStream_t stream) {
  const float* x = (const float*)d_in[0];  // [M, N]
  const float* y = (const float*)d_in[1];  // [M, K]
  const float* z = (const float*)d_in[2];  // [K, N]
  float* out = (float*)d_out;              // [M, N]

  dim3 grid(N_DIM / BLOCK_N, M_DIM / M_CHUNK);  // 32 x 32 blocks
  dim3 block(256);                              // 8 waves (wave32)
  hipLaunchKernelGGL(sddmm_wmma_kernel, grid, block, 0, stream, x, y, z, out);
}